// ConvE_36026185679397
// MI455X (gfx1250) — compile-verified
//
#include <hip/hip_runtime.h>
#include <hip/hip_bf16.h>
#include <math.h>

// ---------------- problem constants ----------------
#define BATCH    1024
#define ENTD     100        // entity embedding dim (10x10)
#define RELD     2500       // relation dim (100 filters * 5*5)
#define NUMENT   100000
#define NFILT    100
#define CONVHW   36         // 6*6
#define CONVLEN  3600       // 100*36
#define EPSV     1e-5f

typedef float v2f __attribute__((ext_vector_type(2)));
typedef float v8f __attribute__((ext_vector_type(8)));

// workspace layout (in floats)
#define OFF_RMEAN    0                         // 2500
#define OFF_RISTD    2500                      // 2500
#define OFF_ESTAT    5000                      // 2 (mean, istd)
#define OFF_CONV     5120                      // 1024*3600
#define OFF_BN1M     (OFF_CONV + BATCH*CONVLEN)        // 100
#define OFF_BN1I     (OFF_BN1M + 128)                  // 100
#define OFF_Y        (OFF_BN1I + 128)                  // 1024*100
#define OFF_BN2M     (OFF_Y + BATCH*ENTD)              // 100
#define OFF_BN2I     (OFF_BN2M + 128)                  // 100
#define OFF_XF       (OFF_BN2I + 128)                  // 1024*100
// total ~3.9M floats (~15.6 MB)

// ---------------------------------------------------------------
// 1) per-channel BN stats of gathered relation embeddings r[b,c]
__global__ void k_rstats(const int* __restrict__ rel,
                         const float* __restrict__ emb_rel,
                         float* __restrict__ ws) {
    int c = blockIdx.x * blockDim.x + threadIdx.x;
    if (c >= RELD) return;
    float s = 0.f, s2 = 0.f;
    for (int b = 0; b < BATCH; ++b) {
        float v = emb_rel[(long)rel[b] * RELD + c];
        s += v; s2 += v * v;
    }
    float mean = s * (1.0f / BATCH);
    float var  = s2 * (1.0f / BATCH) - mean * mean;
    ws[OFF_RMEAN + c] = mean;
    ws[OFF_RISTD + c] = rsqrtf(var + EPSV);
}

// ---------------------------------------------------------------
// 2) single-channel BN stats of gathered entity embeddings (B*100 values)
__global__ void k_estats(const int* __restrict__ e1,
                         const float* __restrict__ emb_e,
                         float* __restrict__ ws) {
    __shared__ float ss[256], ss2[256];
    int tid = threadIdx.x;
    float s = 0.f, s2 = 0.f;
    for (int i = tid; i < BATCH * ENTD; i += 256) {
        int b = i / ENTD, c = i % ENTD;
        float v = emb_e[(long)e1[b] * ENTD + c];
        s += v; s2 += v * v;
    }
    ss[tid] = s; ss2[tid] = s2;
    __syncthreads();
    for (int st = 128; st > 0; st >>= 1) {
        if (tid < st) { ss[tid] += ss[tid + st]; ss2[tid] += ss2[tid + st]; }
        __syncthreads();
    }
    if (tid == 0) {
        float mean = ss[0] / (float)(BATCH * ENTD);
        float var  = ss2[0] / (float)(BATCH * ENTD) - mean * mean;
        ws[OFF_ESTAT + 0] = mean;
        ws[OFF_ESTAT + 1] = rsqrtf(var + EPSV);
    }
}

// ---------------------------------------------------------------
// 3) per-sample dynamic-filter conv: one block per batch element
__global__ void k_conv(const int* __restrict__ e1,
                       const int* __restrict__ rel,
                       const float* __restrict__ emb_e,
                       const float* __restrict__ emb_rel,
                       const float* __restrict__ g0,
                       const float* __restrict__ b0,
                       const float* __restrict__ grel,
                       const float* __restrict__ brel,
                       float* __restrict__ ws) {
    __shared__ float img[ENTD];     // normalized 10x10 image
    __shared__ float filt[RELD];    // normalized 100 filters of 5x5
    int b = blockIdx.x, tid = threadIdx.x;
    float em = ws[OFF_ESTAT + 0], ei = ws[OFF_ESTAT + 1];
    float gg0 = g0[0], bb0 = b0[0];
    for (int i = tid; i < ENTD; i += 256)
        img[i] = (emb_e[(long)e1[b] * ENTD + i] - em) * ei * gg0 + bb0;
    for (int i = tid; i < RELD; i += 256) {
        float v = emb_rel[(long)rel[b] * RELD + i];
        filt[i] = (v - ws[OFF_RMEAN + i]) * ws[OFF_RISTD + i] * grel[i] + brel[i];
    }
    __syncthreads();
    for (int idx = tid; idx < CONVLEN; idx += 256) {
        int f = idx / CONVHW, r = idx % CONVHW;
        int oh = r / 6, ow = r % 6;
        float acc = 0.f;
        const float* fp = &filt[f * 25];
        #pragma unroll
        for (int kh = 0; kh < 5; ++kh)
            #pragma unroll
            for (int kw = 0; kw < 5; ++kw)
                acc = fmaf(img[(oh + kh) * 10 + (ow + kw)], fp[kh * 5 + kw], acc);
        ws[OFF_CONV + (long)b * CONVLEN + idx] = acc;
    }
}

// ---------------------------------------------------------------
// 4) BN1 stats: per filter-channel over (B, 6, 6) = 36864 values
__global__ void k_bn1stats(float* __restrict__ ws) {
    __shared__ float ss[256], ss2[256];
    int f = blockIdx.x, tid = threadIdx.x;
    float s = 0.f, s2 = 0.f;
    for (int j = tid; j < BATCH * CONVHW; j += 256) {
        int b = j / CONVHW, hw = j % CONVHW;
        float v = ws[OFF_CONV + (long)b * CONVLEN + f * CONVHW + hw];
        s += v; s2 += v * v;
    }
    ss[tid] = s; ss2[tid] = s2;
    __syncthreads();
    for (int st = 128; st > 0; st >>= 1) {
        if (tid < st) { ss[tid] += ss[tid + st]; ss2[tid] += ss2[tid + st]; }
        __syncthreads();
    }
    if (tid == 0) {
        float mean = ss[0] / (float)(BATCH * CONVHW);
        float var  = ss2[0] / (float)(BATCH * CONVHW) - mean * mean;
        ws[OFF_BN1M + f] = mean;
        ws[OFF_BN1I + f] = rsqrtf(var + EPSV);
    }
}

// ---------------------------------------------------------------
// 5) FC: y[b,o] = fc_b[o] + sum_j relu(bn1(conv[b,j])) * fc_w[o,j]
__global__ void k_fc(const float* __restrict__ fc_w,
                     const float* __restrict__ fc_b,
                     const float* __restrict__ g1,
                     const float* __restrict__ b1,
                     float* __restrict__ ws) {
    __shared__ float xl[CONVLEN];
    int b = blockIdx.x, tid = threadIdx.x;
    for (int j = tid; j < CONVLEN; j += 128) {
        int f = j / CONVHW;
        float v = (ws[OFF_CONV + (long)b * CONVLEN + j] - ws[OFF_BN1M + f])
                  * ws[OFF_BN1I + f] * g1[f] + b1[f];
        xl[j] = fmaxf(v, 0.f);
    }
    __syncthreads();
    if (tid < ENTD) {
        float acc = fc_b[tid];
        const float* wp = &fc_w[(long)tid * CONVLEN];
        for (int j = 0; j < CONVLEN; ++j)
            acc = fmaf(xl[j], wp[j], acc);
        ws[OFF_Y + (long)b * ENTD + tid] = acc;
    }
}

// ---------------------------------------------------------------
// 6) BN2 stats over batch for each of 100 outputs
__global__ void k_bn2stats(float* __restrict__ ws) {
    __shared__ float ss[256], ss2[256];
    int o = blockIdx.x, tid = threadIdx.x;
    float s = 0.f, s2 = 0.f;
    for (int b = tid; b < BATCH; b += 256) {
        float v = ws[OFF_Y + (long)b * ENTD + o];
        s += v; s2 += v * v;
    }
    ss[tid] = s; ss2[tid] = s2;
    __syncthreads();
    for (int st = 128; st > 0; st >>= 1) {
        if (tid < st) { ss[tid] += ss[tid + st]; ss2[tid] += ss2[tid + st]; }
        __syncthreads();
    }
    if (tid == 0) {
        float mean = ss[0] / (float)BATCH;
        float var  = ss2[0] / (float)BATCH - mean * mean;
        ws[OFF_BN2M + o] = mean;
        ws[OFF_BN2I + o] = rsqrtf(var + EPSV);
    }
}

// 7) apply BN2 + relu -> final activations x_final (B,100)
__global__ void k_bn2apply(const float* __restrict__ g2,
                           const float* __restrict__ b2,
                           float* __restrict__ ws) {
    int i = blockIdx.x * 256 + threadIdx.x;
    if (i >= BATCH * ENTD) return;
    int o = i % ENTD;
    float v = (ws[OFF_Y + i] - ws[OFF_BN2M + o]) * ws[OFF_BN2I + o] * g2[o] + b2[o];
    ws[OFF_XF + i] = fmaxf(v, 0.f);
}

// ---------------------------------------------------------------
// 8) Big GEMM + sigmoid: out[b,n] = sigmoid(x(B,100) . emb_e[n,:] + bias[n])
//    f32 WMMA 16x16x4, K = 100 = 25 steps.
//    - each wave owns TWO 16-col N tiles (32 cols): emb_e frags stay in
//      registers (100 VGPRs) for the whole batch loop; each LDS A-fragment
//      feeds two WMMAs (halves L2 activation traffic vs one tile/wave)
//    - double-buffered LDS activation tile: next tile's global loads issue
//      before the 50-WMMA compute block; one barrier per iteration
//    - NT stores for the 410 MB output stream + NT loads for single-use
//      emb_e rows keep xf L2-resident
#define NPAIRS (NUMENT / 32)   // 3125 (each pair = 32 columns)
__global__ void __launch_bounds__(256)
k_gemm_sigmoid(const float* __restrict__ xf,      // ws + OFF_XF
               const float* __restrict__ emb_e,
               const float* __restrict__ bias,
               float* __restrict__ out) {
    __shared__ float xl[2][16 * ENTD];            // 2 x 6.4 KB
    const int tid  = threadIdx.x;
    const int lane = tid & 31;
    const int pair = blockIdx.x * 8 + (tid >> 5);
    const bool valid = (pair < NPAIRS);
    const int n0 = (valid ? pair : 0) * 32;       // clamp: loads stay legal
    const int n1 = n0 + 16;

    const int m    = lane & 15;                   // row-in-tile (A) / col (B,D)
    const int koff = (lane >> 4) << 1;            // 0 for lanes 0-15, 2 for 16-31

    // preload B fragments (single-use rows -> nontemporal)
    v2f bf0[25], bf1[25];
    {
        const float* p0 = emb_e + (long)(n0 + m) * ENTD + koff;
        const float* p1 = emb_e + (long)(n1 + m) * ENTD + koff;
        #pragma unroll
        for (int kk = 0; kk < 25; ++kk) {
            bf0[kk].x = __builtin_nontemporal_load(p0 + kk * 4 + 0);
            bf0[kk].y = __builtin_nontemporal_load(p0 + kk * 4 + 1);
            bf1[kk].x = __builtin_nontemporal_load(p1 + kk * 4 + 0);
            bf1[kk].y = __builtin_nontemporal_load(p1 + kk * 4 + 1);
        }
    }
    const float bias0 = bias[n0 + m];
    const float bias1 = bias[n1 + m];

    // prologue: stage tile 0 into buffer 0 (800 x float2)
    #pragma unroll
    for (int k = 0; k < 4; ++k) {
        int p = tid + k * 256;
        if (p < 800)
            *(v2f*)&xl[0][p * 2] = *(const v2f*)&xf[p * 2];
    }
    __syncthreads();

    for (int bt = 0; bt < BATCH / 16; ++bt) {
        const int cur = bt & 1, nxt = cur ^ 1;
        const bool more = (bt + 1) < (BATCH / 16);

        // issue next tile's global loads now; latency hides behind WMMAs
        v2f tmp[4];
        if (more) {
            const float* src = xf + (bt + 1) * (16 * ENTD);
            #pragma unroll
            for (int k = 0; k < 4; ++k) {
                int p = tid + k * 256;
                if (p < 800) tmp[k] = *(const v2f*)&src[p * 2];
            }
        }

        if (valid) {
            v8f acc0 = {0.f, 0.f, 0.f, 0.f, 0.f, 0.f, 0.f, 0.f};
            v8f acc1 = {0.f, 0.f, 0.f, 0.f, 0.f, 0.f, 0.f, 0.f};
            const float* xc = &xl[cur][0];
            #pragma unroll
            for (int kk = 0; kk < 25; ++kk) {
                v2f a;
                a.x = xc[m * ENTD + kk * 4 + koff + 0];
                a.y = xc[m * ENTD + kk * 4 + koff + 1];
                acc0 = __builtin_amdgcn_wmma_f32_16x16x4_f32(
                         false, a, false, bf0[kk], (short)0, acc0, false, false);
                acc1 = __builtin_amdgcn_wmma_f32_16x16x4_f32(
                         false, a, false, bf1[kk], (short)0, acc1, false, false);
            }
            const int rbase = bt * 16 + ((lane >> 4) << 3);
            #pragma unroll
            for (int v = 0; v < 8; ++v) {
                float v0 = acc0[v] + bias0;
                float v1 = acc1[v] + bias1;
                long row = (long)(rbase + v) * NUMENT;
                __builtin_nontemporal_store(1.0f / (1.0f + expf(-v0)),
                                            out + row + n0 + m);
                __builtin_nontemporal_store(1.0f / (1.0f + expf(-v1)),
                                            out + row + n1 + m);
            }
        }

        if (more) {
            #pragma unroll
            for (int k = 0; k < 4; ++k) {
                int p = tid + k * 256;
                if (p < 800) *(v2f*)&xl[nxt][p * 2] = tmp[k];
            }
        }
        __syncthreads();
    }
}

// ---------------------------------------------------------------
extern "C" void kernel_launch(void* const* d_in, const int* in_sizes, int n_in,
                              void* d_out, int out_size, void* d_ws, size_t ws_size,
                              hipStream_t stream) {
    const int*   e1      = (const int*)  d_in[0];
    const int*   rel     = (const int*)  d_in[1];
    const float* emb_e   = (const float*)d_in[2];
    const float* emb_rel = (const float*)d_in[3];
    const float* fc_w    = (const float*)d_in[4];
    const float* fc_b    = (const float*)d_in[5];
    const float* bias    = (const float*)d_in[6];
    const float* g0      = (const float*)d_in[7];
    const float* b0      = (const float*)d_in[8];
    const float* g1      = (const float*)d_in[9];
    const float* b1      = (const float*)d_in[10];
    const float* grel    = (const float*)d_in[11];
    const float* brel    = (const float*)d_in[12];
    const float* g2      = (const float*)d_in[13];
    const float* b2      = (const float*)d_in[14];
    float* out = (float*)d_out;
    float* ws  = (float*)d_ws;

    k_rstats  <<<(RELD + 255) / 256, 256, 0, stream>>>(rel, emb_rel, ws);
    k_estats  <<<1, 256, 0, stream>>>(e1, emb_e, ws);
    k_conv    <<<BATCH, 256, 0, stream>>>(e1, rel, emb_e, emb_rel,
                                          g0, b0, grel, brel, ws);
    k_bn1stats<<<NFILT, 256, 0, stream>>>(ws);
    k_fc      <<<BATCH, 128, 0, stream>>>(fc_w, fc_b, g1, b1, ws);
    k_bn2stats<<<ENTD, 256, 0, stream>>>(ws);
    k_bn2apply<<<(BATCH * ENTD + 255) / 256, 256, 0, stream>>>(g2, b2, ws);
    k_gemm_sigmoid<<<(NPAIRS + 7) / 8, 256, 0, stream>>>(ws + OFF_XF, emb_e,
                                                         bias, out);
}